// SKEL_40312563040784
// MI455X (gfx1250) — compile-verified
//
#include <hip/hip_runtime.h>
#include <hip/hip_bf16.h>

typedef float v2f __attribute__((ext_vector_type(2)));
typedef float v8f __attribute__((ext_vector_type(8)));

#define NB 512
#define NV 6890
#define NJ 24
#define KF 220          // padded features: 10 betas + 207 pose_feat + 1 const + 2 pad
#define REC 12          // per-joint record: 9 GR + 3 (A_t + trans)

// workspace layout (float offsets)
#define OFF_F   0                        // [KF][NB]
#define OFF_RA  (KF*NB)                  // [NB][NJ][REC]
#define OFF_JS  (OFF_RA + NB*NJ*REC)     // [NJ][3][10]
#define OFF_JT  (OFF_JS + NJ*3*10)       // [NJ][3]

// ---------- baked model constants (from reference _build_axes) ----------
__device__ constexpr float c_axes[47*3] = {
  0,0,1,   1,0,0,   0,1,0,            // j0
  0,0,1,   1,0,0,   0,1,0,            // j1
  0,0,-1,                             // j2
 -0.1050139f,-0.1740218f, 0.9791266f, // j3 pin
  0.7871802f, 0.6047471f,-0.1209489f, // j4 pin
  0.5809549f, 0.0000022f,-0.8139361f, // j5 pin
  0,0,1,  -1,0,0,   0,-1,0,           // j6
  0,0,-1,                             // j7
 -0.1050139f,-0.1740218f, 0.9791266f, // j8 pin
 -0.7871802f,-0.6047471f,-0.1209489f, // j9 pin
 -0.5809549f, 0.0000022f,-0.8139361f, // j10 pin
  1,0,0,   0,0,1,   0,1,0,            // j11
  1,0,0,   0,0,1,   0,1,0,            // j12
  1,0,0,   0,0,1,   0,1,0,            // j13
  0,1,0,   0,0,-1, -1,0,0,            // j14
  1,0,0,   0,1,0,   0,0,1,            // j15
  0.0494f, 0.0366f, 0.99810825f,      // j16
 -0.01716099f, 0.99266564f,-0.11966796f, // j17
  1,0,0,   0,0,-1,                    // j18
  0,1,0,   0,0,1,   1,0,0,            // j19
  1,0,0,   0,1,0,   0,0,1,            // j20
 -0.0494f,-0.0366f, 0.99810825f,      // j21
  0.01716099f,-0.99266564f,-0.11966796f, // j22
 -1,0,0,  0,0,-1,                     // j23
  0,0,0                               // idx 46: padded identity DOF
};

__device__ constexpr int c_ridx[NJ*3] = {
  0,1,2,   3,4,5,   6,46,46, 7,46,46, 8,46,46, 9,46,46,
  10,11,12, 13,46,46, 14,46,46, 15,46,46, 16,46,46,
  17,18,19, 20,21,22, 23,24,25, 26,27,28, 29,30,31,
  32,46,46, 33,46,46, 34,35,46, 36,37,38, 39,40,41,
  42,46,46, 43,46,46, 44,45,46
};

__device__ constexpr int c_par[NJ] = {
  -1,0,1,2,3,4,0,6,7,8,9,0,11,12,12,14,15,16,17,12,19,20,21,22
};

// ---------- small mat3 helpers ----------
__device__ __forceinline__ void m3mul(const float* A, const float* B, float* C) {
#pragma unroll
  for (int m = 0; m < 3; ++m)
#pragma unroll
    for (int n = 0; n < 3; ++n)
      C[m*3+n] = A[m*3+0]*B[0*3+n] + A[m*3+1]*B[1*3+n] + A[m*3+2]*B[2*3+n];
}
__device__ __forceinline__ void m3mulBT(const float* A, const float* B, float* C) {
#pragma unroll
  for (int m = 0; m < 3; ++m)
#pragma unroll
    for (int n = 0; n < 3; ++n)
      C[m*3+n] = A[m*3+0]*B[n*3+0] + A[m*3+1]*B[n*3+1] + A[m*3+2]*B[n*3+2];
}
__device__ __forceinline__ void rodrigues_unit(float q, float ax, float ay, float az, float* R) {
  float s = __sinf(q), c = __cosf(q), t = 1.f - c;
  R[0] = c + t*ax*ax;      R[1] = t*ax*ay - s*az;  R[2] = t*ax*az + s*ay;
  R[3] = t*ax*ay + s*az;   R[4] = c + t*ay*ay;     R[5] = t*ay*az - s*ax;
  R[6] = t*ax*az - s*ay;   R[7] = t*ay*az + s*ax;  R[8] = c + t*az*az;
}

// ---------- Kernel A: JS = Jr . shapedirs, JT = Jr . v_template ----------
// grid (33, 24), 256 threads; idx<30 -> JS[c=idx/10][k=idx%10], idx>=30 -> JT[idx-30]
__global__ __launch_bounds__(256) void skel_jreg_kernel(
    const float* __restrict__ Jr, const float* __restrict__ sd,
    const float* __restrict__ vt, float* __restrict__ JS, float* __restrict__ JT)
{
  __shared__ float red[256];
  const int j = blockIdx.y, idx = blockIdx.x, t = threadIdx.x;
  float s = 0.f;
  if (idx < 30) {
    const int c = idx / 10, k = idx % 10;
    for (int v = t; v < NV; v += 256)
      s += Jr[(size_t)j*NV + v] * sd[((size_t)v*3 + c)*10 + k];
  } else {
    const int c = idx - 30;
    for (int v = t; v < NV; v += 256)
      s += Jr[(size_t)j*NV + v] * vt[(size_t)v*3 + c];
  }
  red[t] = s;
  for (int o = 128; o > 0; o >>= 1) {
    __syncthreads();
    if (t < o) red[t] += red[t + o];
  }
  if (t == 0) {
    if (idx < 30) JS[(j*3 + idx/10)*10 + idx%10] = red[0];
    else          JT[j*3 + (idx - 30)] = red[0];
  }
}

// ---------- Kernel B: per-batch rotations, FK, feature matrix, GR/At records ----------
__global__ __launch_bounds__(256) void skel_prep_kernel(
    const float* __restrict__ betas, const float* __restrict__ poses,
    const float* __restrict__ trans, const float* __restrict__ aposeR,
    const float* __restrict__ pjr,   const float* __restrict__ JS,
    const float* __restrict__ JT,    float* __restrict__ F,
    float* __restrict__ RA)
{
  const int b = blockIdx.x * 256 + threadIdx.x;
  if (b >= NB) return;

  // joint locations: Jl = JT + JS @ betas[b]
  float bb[10];
#pragma unroll
  for (int k = 0; k < 10; ++k) bb[k] = betas[(size_t)b*10 + k];
  float Jl[NJ][3];
  for (int j = 0; j < NJ; ++j)
#pragma unroll
    for (int c = 0; c < 3; ++c) {
      float s = JT[j*3 + c];
      const float* js = JS + (j*3 + c)*10;
#pragma unroll
      for (int k = 0; k < 10; ++k) s = fmaf(js[k], bb[k], s);
      Jl[j][c] = s;
    }

  float GR[NJ][9], Gt[NJ][3];
  const float* pb = poses + (size_t)b*46;

  for (int j = 0; j < NJ; ++j) {
    // Rj = Rd[r0] @ Rd[r1] @ Rd[r2]
    float Rj[9] = {1,0,0, 0,1,0, 0,0,1};
#pragma unroll
    for (int s3 = 0; s3 < 3; ++s3) {
      const int d = c_ridx[j*3 + s3];
      if (d == 46) break;               // padded identity DOF
      float R[9], tmp[9];
      rodrigues_unit(pb[d], c_axes[d*3], c_axes[d*3+1], c_axes[d*3+2], R);
      m3mul(Rj, R, tmp);
#pragma unroll
      for (int i = 0; i < 9; ++i) Rj[i] = tmp[i];
    }
    // pose features (Rj[1:] - I) into F rows 10..216, layout F[k][b]
    if (j > 0) {
      const int base = 10 + (j - 1) * 9;
#pragma unroll
      for (int i = 0; i < 9; ++i) {
        float e = Rj[i] - ((i == 0 || i == 4 || i == 8) ? 1.f : 0.f);
        F[(size_t)(base + i)*NB + b] = e;
      }
    }
    // R_local = apose_R @ (P @ Rj @ P^T)
    const float* P = pjr + j*9;
    const float* A = aposeR + j*9;
    float t1[9], t2[9], RL[9];
    m3mul(P, Rj, t1);
    m3mulBT(t1, P, t2);
    m3mul(A, t2, RL);
    // forward kinematics
    const int p = c_par[j];
    if (p < 0) {
#pragma unroll
      for (int i = 0; i < 9; ++i) GR[0][i] = RL[i];
#pragma unroll
      for (int c = 0; c < 3; ++c) Gt[0][c] = Jl[0][c];
    } else {
      m3mul(GR[p], RL, GR[j]);
      float d0 = Jl[j][0]-Jl[p][0], d1 = Jl[j][1]-Jl[p][1], d2 = Jl[j][2]-Jl[p][2];
#pragma unroll
      for (int m = 0; m < 3; ++m)
        Gt[j][m] = Gt[p][m] + GR[p][m*3]*d0 + GR[p][m*3+1]*d1 + GR[p][m*3+2]*d2;
    }
  }

  // betas + const + pad rows of F
#pragma unroll
  for (int k = 0; k < 10; ++k) F[(size_t)k*NB + b] = bb[k];
  F[(size_t)217*NB + b] = 1.f;
  F[(size_t)218*NB + b] = 0.f;
  F[(size_t)219*NB + b] = 0.f;

  // records: [GR(9) | Gt - GR@Jl + trans(3)]
  const float tr0 = trans[(size_t)b*3], tr1 = trans[(size_t)b*3+1], tr2 = trans[(size_t)b*3+2];
  for (int j = 0; j < NJ; ++j) {
    float* rec = RA + ((size_t)b*NJ + j)*REC;
#pragma unroll
    for (int i = 0; i < 9; ++i) rec[i] = GR[j][i];
    float tcm[3] = {tr0, tr1, tr2};
#pragma unroll
    for (int m = 0; m < 3; ++m)
      rec[9+m] = Gt[j][m]
               - (GR[j][m*3]*Jl[j][0] + GR[j][m*3+1]*Jl[j][1] + GR[j][m*3+2]*Jl[j][2])
               + tcm[m];
  }
}

// ---------- Kernel C: WMMA blend-shape GEMM + fused LBS skinning ----------
// grid (ceil(V/16)=431, B/16=32), one wave (32 threads) per block.
// Tile: 16 batches x 16 vertices x 3 components; K loop over 220 features via
// V_WMMA_F32_16X16X4_F32 (A: F[220][512] K-major, B: shapedirs/posedirs/v_template rows).
__global__ __launch_bounds__(32) void skel_lbs_kernel(
    const float* __restrict__ F,  const float* __restrict__ RA,
    const float* __restrict__ sd, const float* __restrict__ pd,
    const float* __restrict__ vt, const float* __restrict__ sw,
    float* __restrict__ out)
{
  __shared__ __align__(16) float ra[16*NJ*REC];   // 18432 B: GR/At records for 16 batches
  const int ln    = threadIdx.x;
  const int vbase = blockIdx.x * 16;
  const int bbase = blockIdx.y * 16;

  // stage records for this batch tile into LDS (coalesced b128)
  {
    const float4* src = (const float4*)(RA + (size_t)bbase*NJ*REC);
    float4* dst = (float4*)ra;
#pragma unroll 4
    for (int i = ln; i < (16*NJ*REC)/4; i += 32) dst[i] = src[i];
  }
  __syncthreads();

  const int  hi    = ln >> 4;       // half-wave selects K pair (+0/+1 vs +2/+3)
  const int  col   = ln & 15;
  const int  vg    = vbase + col;
  const bool valid = (vg < NV);
  const int  vc    = valid ? vg : (NV - 1);

  // A fragment source: F is [K][B]; this lane reads rows (k+2*hi, k+2*hi+1), col bbase+col
  const float* F0   = F + (size_t)(2*hi)*NB + bbase + col;
  const float* sdp0 = sd + (size_t)vc * 30;     // [v][c][10]
  const float* pdp0 = pd + (size_t)vc * 3 * 207; // [v][c][207]

  v8f accx = {}, accy = {}, accz = {};

  for (int k = 0; k < 216; k += 4) {
    const int k0 = k + 2*hi;
    v2f a;
    a.x = F0[(size_t)k*NB];
    a.y = F0[(size_t)k*NB + NB];
    v2f b0, b1, b2;
    if (k0 < 10) {                         // shape blend segment (pairs stay inside)
      b0 = *(const v2f*)(sdp0 + k0);
      b1 = *(const v2f*)(sdp0 + 10 + k0);
      b2 = *(const v2f*)(sdp0 + 20 + k0);
    } else {                               // pose blend segment
      const float* p0 = pdp0 + (k0 - 10);
      b0.x = p0[0];    b0.y = p0[1];
      b1.x = p0[207];  b1.y = p0[208];
      b2.x = p0[414];  b2.y = p0[415];
    }
    accx = __builtin_amdgcn_wmma_f32_16x16x4_f32(false, a, false, b0, (short)0, accx, false, false);
    accy = __builtin_amdgcn_wmma_f32_16x16x4_f32(false, a, false, b1, (short)0, accy, false, false);
    accz = __builtin_amdgcn_wmma_f32_16x16x4_f32(false, a, false, b2, (short)0, accz, false, false);
  }

  // tail step k=216: K pair (216,217) = (posedirs[206], v_template); (218,219) = zero pad
  {
    v2f a;
    a.x = F0[(size_t)216*NB];
    a.y = F0[(size_t)216*NB + NB];
    v2f b0 = {0.f, 0.f}, b1 = {0.f, 0.f}, b2 = {0.f, 0.f};
    if (hi == 0) {
      b0.x = pdp0[206];        b0.y = vt[(size_t)vc*3 + 0];
      b1.x = pdp0[207 + 206];  b1.y = vt[(size_t)vc*3 + 1];
      b2.x = pdp0[414 + 206];  b2.y = vt[(size_t)vc*3 + 2];
    }
    accx = __builtin_amdgcn_wmma_f32_16x16x4_f32(false, a, false, b0, (short)0, accx, false, false);
    accy = __builtin_amdgcn_wmma_f32_16x16x4_f32(false, a, false, b1, (short)0, accy, false, false);
    accz = __builtin_amdgcn_wmma_f32_16x16x4_f32(false, a, false, b2, (short)0, accz, false, false);
  }

  // epilogue: linear blend skinning. Lane holds vertex vc, batch rows g+8*hi.
  float w[NJ];
#pragma unroll
  for (int j = 0; j < NJ; ++j) w[j] = sw[(size_t)vc*NJ + j];

#pragma unroll
  for (int g = 0; g < 8; ++g) {
    const int brow = g + 8*hi;
    const float vx = accx[g], vy = accy[g], vz = accz[g];
    float o0 = 0.f, o1 = 0.f, o2 = 0.f;
    const float* rec = ra + brow*NJ*REC;
    for (int j = 0; j < NJ; ++j) {
      const float4 r0 = *(const float4*)(rec + j*REC);
      const float4 r1 = *(const float4*)(rec + j*REC + 4);
      const float4 r2 = *(const float4*)(rec + j*REC + 8);
      const float y0 = fmaf(r0.x, vx, fmaf(r0.y, vy, fmaf(r0.z, vz, r2.y)));
      const float y1 = fmaf(r0.w, vx, fmaf(r1.x, vy, fmaf(r1.y, vz, r2.z)));
      const float y2 = fmaf(r1.z, vx, fmaf(r1.w, vy, fmaf(r2.x, vz, r2.w)));
      const float wj = w[j];
      o0 = fmaf(wj, y0, o0);
      o1 = fmaf(wj, y1, o1);
      o2 = fmaf(wj, y2, o2);
    }
    if (valid) {
      float* o = out + ((size_t)(bbase + brow)*NV + vg)*3;
      o[0] = o0; o[1] = o1; o[2] = o2;
    }
  }
}

// ---------- launcher ----------
extern "C" void kernel_launch(void* const* d_in, const int* in_sizes, int n_in,
                              void* d_out, int out_size, void* d_ws, size_t ws_size,
                              hipStream_t stream) {
  const float* betas = (const float*)d_in[0];
  const float* poses = (const float*)d_in[1];
  const float* trans = (const float*)d_in[2];
  const float* vtem  = (const float*)d_in[3];
  const float* sd    = (const float*)d_in[4];
  const float* pd    = (const float*)d_in[5];
  const float* Jr    = (const float*)d_in[6];
  const float* sw    = (const float*)d_in[7];
  const float* apR   = (const float*)d_in[8];
  const float* pjr   = (const float*)d_in[9];
  float* out = (float*)d_out;
  float* ws  = (float*)d_ws;

  skel_jreg_kernel<<<dim3(33, 24), 256, 0, stream>>>(Jr, sd, vtem, ws + OFF_JS, ws + OFF_JT);
  skel_prep_kernel<<<dim3(2), 256, 0, stream>>>(betas, poses, trans, apR, pjr,
                                                ws + OFF_JS, ws + OFF_JT,
                                                ws + OFF_F, ws + OFF_RA);
  skel_lbs_kernel<<<dim3((NV + 15) / 16, NB / 16), 32, 0, stream>>>(
      ws + OFF_F, ws + OFF_RA, sd, pd, vtem, sw, out);
}